// RetinaNet_45122926412096
// MI455X (gfx1250) — compile-verified
//
#include <hip/hip_runtime.h>
#include <stdint.h>

#define NEGV      (-1000000000.0f)
#define NEG_HALF  (-500000000.0f)
#define PST_THD_F (0.05f)
#define NMS_THD_F (0.5f)
#define MAX_DET   300
#define CHUNK     512
#define MAXCH     512   // supports A up to 262144 anchors

// ---------------- wave32 reduction helpers ----------------
__device__ __forceinline__ float waveMaxF(float v) {
    for (int o = 16; o; o >>= 1) v = fmaxf(v, __shfl_xor(v, o, 32));
    return v;
}
__device__ __forceinline__ float waveMinF(float v) {
    for (int o = 16; o; o >>= 1) v = fminf(v, __shfl_xor(v, o, 32));
    return v;
}
__device__ __forceinline__ void waveArgmax(float& v, int& idx) {
    for (int o = 16; o; o >>= 1) {
        float ov = __shfl_xor(v, o, 32);
        int   oi = __shfl_xor(idx, o, 32);
        if (ov > v || (ov == v && oi < idx)) { v = ov; idx = oi; }
    }
}

// ---------------------------------------------------------------------------
// Kernel 1: per-anchor class max, box decode+clip, area, candidate score.
// Bandwidth bound: 63MB classification read once (~2.7us at 23.3 TB/s).
// ---------------------------------------------------------------------------
__global__ void decode_score_kernel(const float* __restrict__ cls,
                                    const float* __restrict__ reg,
                                    const float* __restrict__ anc,
                                    const int* __restrict__ ih,
                                    const int* __restrict__ iw,
                                    int A, int C,
                                    float4* __restrict__ boxes,
                                    float* __restrict__ areas,
                                    float* __restrict__ cand) {
    int a = blockIdx.x * blockDim.x + threadIdx.x;
    if (a >= A) return;
    float W = (float)iw[0], H = (float)ih[0];

    float4 an = ((const float4*)anc)[a];
    float4 rg = ((const float4*)reg)[a];
    float w  = an.z - an.x, h = an.w - an.y;
    float cx = an.x + 0.5f * w, cy = an.y + 0.5f * h;
    float pcx = cx + rg.x * 0.1f * w;
    float pcy = cy + rg.y * 0.1f * h;
    float pw  = expf(rg.z * 0.2f) * w;
    float ph  = expf(rg.w * 0.2f) * h;
    float x1 = fmaxf(pcx - 0.5f * pw, 0.0f);
    float y1 = fmaxf(pcy - 0.5f * ph, 0.0f);
    float x2 = fminf(pcx + 0.5f * pw, W);
    float y2 = fminf(pcy + 0.5f * ph, H);
    boxes[a] = make_float4(x1, y1, x2, y2);
    areas[a] = (x2 - x1) * (y2 - y1);

    // max over C classes, vectorized b128 loads on contiguous row
    const float4* row = (const float4*)(cls + (size_t)a * C);
    float m = -1e30f;
    int nv = C >> 2;
    for (int k = 0; k < nv; k++) {
        float4 v = row[k];
        m = fmaxf(m, fmaxf(fmaxf(v.x, v.y), fmaxf(v.z, v.w)));
    }
    for (int k = nv << 2; k < C; k++) m = fmaxf(m, cls[(size_t)a * C + k]);
    cand[a] = (m > PST_THD_F) ? m : NEGV;
}

// ---------------------------------------------------------------------------
// Kernel 2: per-chunk metadata: [cmax, bx1, by1, bx2, by2, amin, amax, pad]
// ---------------------------------------------------------------------------
__global__ void chunk_meta_kernel(const float4* __restrict__ boxes,
                                  const float* __restrict__ areas,
                                  const float* __restrict__ cand,
                                  int A, float* __restrict__ meta) {
    int c = blockIdx.x;
    int base = c * CHUNK;
    float cmax = NEGV, bx1 = 1e30f, by1 = 1e30f, bx2 = -1e30f, by2 = -1e30f;
    float amn = 1e30f, amx = -1e30f;
    for (int e = threadIdx.x; e < CHUNK; e += blockDim.x) {
        int i = base + e;
        if (i < A) {
            cmax = fmaxf(cmax, cand[i]);
            float4 b = boxes[i];
            bx1 = fminf(bx1, b.x); by1 = fminf(by1, b.y);
            bx2 = fmaxf(bx2, b.z); by2 = fmaxf(by2, b.w);
            float ar = areas[i];
            amn = fminf(amn, ar);  amx = fmaxf(amx, ar);
        }
    }
    __shared__ float red[7][8];
    int lane = threadIdx.x & 31, wid = threadIdx.x >> 5;
    cmax = waveMaxF(cmax); bx1 = waveMinF(bx1); by1 = waveMinF(by1);
    bx2 = waveMaxF(bx2);   by2 = waveMaxF(by2);
    amn = waveMinF(amn);   amx = waveMaxF(amx);
    if (lane == 0) {
        red[0][wid] = cmax; red[1][wid] = bx1; red[2][wid] = by1;
        red[3][wid] = bx2;  red[4][wid] = by2; red[5][wid] = amn; red[6][wid] = amx;
    }
    __syncthreads();
    if (threadIdx.x == 0) {
        int nw = blockDim.x >> 5;
        float r0 = NEGV, r1 = 1e30f, r2 = 1e30f, r3 = -1e30f, r4 = -1e30f, r5 = 1e30f, r6 = -1e30f;
        for (int k = 0; k < nw; k++) {
            r0 = fmaxf(r0, red[0][k]); r1 = fminf(r1, red[1][k]); r2 = fminf(r2, red[2][k]);
            r3 = fmaxf(r3, red[3][k]); r4 = fmaxf(r4, red[4][k]);
            r5 = fminf(r5, red[5][k]); r6 = fmaxf(r6, red[6][k]);
        }
        float* m = meta + (size_t)c * 8;
        m[0] = r0; m[1] = r1; m[2] = r2; m[3] = r3; m[4] = r4; m[5] = r5; m[6] = r6; m[7] = 0.0f;
    }
}

// ---------------------------------------------------------------------------
// Kernel 3: persistent single-workgroup greedy NMS (exact reference semantics)
// Chunk metadata is LDS-resident; winning chunk is staged into LDS via the
// gfx1250 async-to-LDS copy path; suppression prunes chunks geometrically.
// ---------------------------------------------------------------------------
__global__ __launch_bounds__(1024, 1)
void nms_kernel(const float4* __restrict__ boxes,
                const float* __restrict__ areas,
                float* __restrict__ cand,
                const float* __restrict__ meta,
                int A, int nChunks,
                int* __restrict__ keep, int* __restrict__ valid) {
    const int tid = threadIdx.x;
    const int lane = tid & 31, wid = tid >> 5, nw = blockDim.x >> 5;

    __shared__ float sMeta[MAXCH * 8];
    __shared__ float sChunk[CHUNK];
    __shared__ float sV[32];
    __shared__ int   sI[32];
    __shared__ float sBox[5];   // jx1,jy1,jx2,jy2,aj

    for (int k = tid; k < nChunks * 8; k += blockDim.x) sMeta[k] = meta[k];
    __syncthreads();

    int i;
    for (i = 0; i < MAX_DET; i++) {
        // ---- phase 1: argmax over LDS chunk maxima ----
        float v = -2e9f; int ci = 0x7fffffff;
        if (tid < nChunks) { v = sMeta[tid * 8 + 0]; ci = tid; }
        waveArgmax(v, ci);
        if (lane == 0) { sV[wid] = v; sI[wid] = ci; }
        __syncthreads();
        if (wid == 0) {
            float vv = (lane < nw) ? sV[lane] : -1e30f;
            int   ii = (lane < nw) ? sI[lane] : 0x7fffffff;
            waveArgmax(vv, ii);
            if (lane == 0) { sV[0] = vv; sI[0] = ii; }
        }
        __syncthreads();
        int cbest = sI[0];
        __syncthreads();

        // ---- phase 2: argmax inside winning chunk; async-stage to LDS ----
        int base  = cbest * CHUNK;
        int count = min(CHUNK, A - base);
        if (tid < count) {
            uint32_t ldsoff = (uint32_t)(uintptr_t)(&sChunk[tid]);
            uint32_t voff   = (uint32_t)tid * 4u;
            uint64_t sa     = (uint64_t)(uintptr_t)(cand + base);
            asm volatile("global_load_async_to_lds_b32 %0, %1, %2"
                         :: "v"(ldsoff), "v"(voff), "s"(sa) : "memory");
        }
        asm volatile("s_wait_asynccnt 0" ::: "memory");
        __syncthreads();

        float v2 = -2e9f; int i2 = 0x7fffffff;
        if (tid < count) { v2 = sChunk[tid]; i2 = base + tid; }
        waveArgmax(v2, i2);
        if (lane == 0) { sV[wid] = v2; sI[wid] = i2; }
        __syncthreads();
        if (wid == 0) {
            float vv = (lane < nw) ? sV[lane] : -1e30f;
            int   ii = (lane < nw) ? sI[lane] : 0x7fffffff;
            waveArgmax(vv, ii);
            if (lane == 0) { sV[0] = vv; sI[0] = ii; }
        }
        __syncthreads();
        int   j  = sI[0];
        float sj = sV[0];
        __syncthreads();

        bool ok = (sj > NEG_HALF);
        if (tid == 0) { keep[i] = j; valid[i] = ok ? 1 : 0; }
        if (!ok) break;   // all remaining entries are exactly NEG -> keep=0, valid=false

        if (tid == 0) {
            float4 bj = boxes[j];
            sBox[0] = bj.x; sBox[1] = bj.y; sBox[2] = bj.z; sBox[3] = bj.w;
            sBox[4] = areas[j];
        }
        __syncthreads();
        float jx1 = sBox[0], jy1 = sBox[1], jx2 = sBox[2], jy2 = sBox[3], aj = sBox[4];

        // ---- phase 3: pruned suppression; one wave per chunk ----
        for (int c = wid; c < nChunks; c += nw) {
            float cm = sMeta[c * 8 + 0];
            if (cm <= NEG_HALF) continue;                       // chunk fully dead
            float mx1 = sMeta[c * 8 + 1], my1 = sMeta[c * 8 + 2];
            float mx2 = sMeta[c * 8 + 3], my2 = sMeta[c * 8 + 4];
            if (mx1 >= jx2 || mx2 <= jx1 || my1 >= jy2 || my2 <= jy1) continue; // no overlap
            float amn = sMeta[c * 8 + 5], amx = sMeta[c * 8 + 6];
            if (amx <= 0.5f * aj || amn >= 2.0f * aj) continue; // IoU>0.5 impossible
            int cb = c * CHUNK;
            int cc = min(CHUNK, A - cb);
            __builtin_prefetch(&boxes[cb], 0, 0);               // global_prefetch_b8
            float nm = NEGV;
            for (int e = lane; e < cc; e += 32) {
                int k = cb + e;
                float s = cand[k];
                if (s > NEG_HALF) {
                    float4 bk = boxes[k];
                    float xx1 = fmaxf(jx1, bk.x), yy1 = fmaxf(jy1, bk.y);
                    float xx2 = fminf(jx2, bk.z), yy2 = fminf(jy2, bk.w);
                    float inter = fmaxf(xx2 - xx1, 0.0f) * fmaxf(yy2 - yy1, 0.0f);
                    float iou = inter / (areas[k] + aj - inter + 1e-8f);
                    if (iou > NMS_THD_F) { cand[k] = NEGV; s = NEGV; }
                }
                nm = fmaxf(nm, s);
            }
            nm = waveMaxF(nm);
            if (lane == 0) sMeta[c * 8 + 0] = nm;               // refresh chunk max
        }
        __syncthreads();
    }
    // fill remaining detections (exhausted state): keep=0, valid=false
    for (int r = i + 1 + tid; r < MAX_DET; r += blockDim.x) { keep[r] = 0; valid[r] = 0; }
}

// ---------------------------------------------------------------------------
// Kernel 4: one wave per kept detection -> class max/argmax + outputs
// out layout (float): scores[300] | class[300] | boxes[300*4] | valid[300]
// ---------------------------------------------------------------------------
__global__ void finalize_kernel(const float* __restrict__ cls,
                                const float4* __restrict__ boxes,
                                const int* __restrict__ keep,
                                const int* __restrict__ valid,
                                int C, float* __restrict__ out) {
    int gw = (blockIdx.x * blockDim.x + threadIdx.x) >> 5;
    int lane = threadIdx.x & 31;
    if (gw >= MAX_DET) return;
    int j = keep[gw];
    int ok = valid[gw];
    const float* row = cls + (size_t)j * C;
    float m = -1e30f; int am = 0x7fffffff;
    for (int k = lane; k < C; k += 32) {
        float s = row[k];
        if (s > m || (s == m && k < am)) { m = s; am = k; }
    }
    waveArgmax(m, am);
    float* out_scores = out;
    float* out_class  = out + MAX_DET;
    float* out_boxes  = out + 2 * MAX_DET;
    float* out_valid  = out + 6 * MAX_DET;
    if (lane == 0) {
        out_scores[gw] = ok ? m : 0.0f;
        out_class[gw]  = ok ? (float)am : -1.0f;
        out_valid[gw]  = ok ? 1.0f : 0.0f;
    }
    if (lane < 4) {
        const float* b = (const float*)(boxes + j);
        out_boxes[gw * 4 + lane] = ok ? b[lane] : 0.0f;
    }
}

// ---------------------------------------------------------------------------
extern "C" void kernel_launch(void* const* d_in, const int* in_sizes, int n_in,
                              void* d_out, int out_size, void* d_ws, size_t ws_size,
                              hipStream_t stream) {
    const float* cls = (const float*)d_in[0];
    const float* reg = (const float*)d_in[1];
    const float* anc = (const float*)d_in[2];
    const int*   ih  = (const int*)d_in[3];
    const int*   iw  = (const int*)d_in[4];

    int A = in_sizes[1] / 4;          // regression is (1, A, 4)
    int C = in_sizes[0] / A;          // classification is (1, A, C)
    int nChunks = (A + CHUNK - 1) / CHUNK;

    // workspace layout (~4.8 MB for A=196416)
    char* p = (char*)d_ws;
    float4* boxes = (float4*)p;  p += (size_t)A * sizeof(float4);
    float*  areas = (float*)p;   p += (size_t)A * sizeof(float);
    float*  cand  = (float*)p;   p += (size_t)A * sizeof(float);
    float*  meta  = (float*)p;   p += (size_t)nChunks * 8 * sizeof(float);
    int*    keep  = (int*)p;     p += (size_t)MAX_DET * sizeof(int);
    int*    valid = (int*)p;

    decode_score_kernel<<<(A + 255) / 256, 256, 0, stream>>>(
        cls, reg, anc, ih, iw, A, C, boxes, areas, cand);
    chunk_meta_kernel<<<nChunks, 256, 0, stream>>>(boxes, areas, cand, A, meta);
    nms_kernel<<<1, 1024, 0, stream>>>(boxes, areas, cand, meta, A, nChunks, keep, valid);
    finalize_kernel<<<(MAX_DET * 32 + 255) / 256, 256, 0, stream>>>(
        cls, boxes, keep, valid, C, (float*)d_out);
}